// Encoder2_25031069401691
// MI455X (gfx1250) — compile-verified
//
#include <hip/hip_runtime.h>
#include <hip/hip_bf16.h>

typedef __attribute__((ext_vector_type(2))) float v2f;
typedef __attribute__((ext_vector_type(8))) float v8f;

#define D 128

// ---------------------------------------------------------------------------
// Kernel 0: zero the aggregation buffer + column-sum slots (grid-stride).
// ---------------------------------------------------------------------------
__global__ __launch_bounds__(256) void zero_f32(float* __restrict__ p, size_t n) {
    size_t stride = (size_t)gridDim.x * blockDim.x;
    for (size_t i = (size_t)blockIdx.x * blockDim.x + threadIdx.x; i < n; i += stride)
        p[i] = 0.0f;
}

// ---------------------------------------------------------------------------
// Kernel 1: edge scatter.  One wave (32 lanes) per edge; each lane owns a
// float4 chunk of the 128-wide feature row.  Loads are fully coalesced
// (512 B / wave); scatter is non-returning global_atomic_add_f32 into agg,
// which is L2-resident (51 MB < 192 MB L2).
// ---------------------------------------------------------------------------
__global__ __launch_bounds__(256) void edge_scatter(
    const float* __restrict__ feat, const int* __restrict__ src,
    const int* __restrict__ dst, const float* __restrict__ ew,
    float* __restrict__ agg, int E)
{
    long long gid = (long long)blockIdx.x * blockDim.x + threadIdx.x;
    int e = (int)(gid >> 5);
    if (e >= E) return;
    int lane = (int)(gid & 31);

    int s = src[e];
    int d = dst[e];
    float w = ew[e];

    const float4 f = ((const float4*)(feat + (size_t)s * D))[lane];
    float* ap = agg + (size_t)d * D + lane * 4;
    atomicAdd(ap + 0, f.x * w);
    atomicAdd(ap + 1, f.y * w);
    atomicAdd(ap + 2, f.z * w);
    atomicAdd(ap + 3, f.w * w);
}

// ---------------------------------------------------------------------------
// Kernel 2: fp32 WMMA GEMM  h = prelu(agg @ W + b, a1)  + BN partial sums.
// One wave computes one 16x16 output tile via 32x V_WMMA_F32_16X16X4_F32,
// split into TWO independent accumulator chains (even/odd K-quads) so the
// scheduler can overlap dependent-WMMA latency instead of serializing all 32.
// 8 waves/block cover one 16-row stripe across all 128 columns.
//
// f32 16x16x4 fragment layouts (ISA 7.12.2):
//   A (16x4):  lanes 0-15 hold K=0,1 (vgpr0,1); lanes 16-31 hold K=2,3
//   B (4x16):  vgpr0 = rows K=0 (lanes 0-15) / K=2 (lanes 16-31); vgpr1 = K=1/K=3
//   C/D:       vgpr v = row m0+v (lanes 0-15) / row m0+v+8 (lanes 16-31),
//              column j0+lane%16  -> each lane's 8 accs share one column,
//              which makes the per-column BN reduction free.
// ---------------------------------------------------------------------------
__global__ __launch_bounds__(256) void gemm_wmma(
    const float* __restrict__ agg, const float* __restrict__ W,
    const float* __restrict__ bias, const float* __restrict__ a1p,
    float* __restrict__ h_out, float* __restrict__ colsum,
    float* __restrict__ colsumsq)
{
    __shared__ float s_sum[D];
    __shared__ float s_sq[D];
    int tid = threadIdx.x;
    if (tid < D) { s_sum[tid] = 0.0f; s_sq[tid] = 0.0f; }
    __syncthreads();

    const int wave = tid >> 5;       // 0..7 -> column tile
    const int lane = tid & 31;
    const int half = lane >> 4;      // 0: lanes 0-15, 1: lanes 16-31
    const int l    = lane & 15;
    const int m0   = blockIdx.x * 16;
    const int j0   = wave * 16;
    const float a1 = a1p[0];

    const float* Arow = agg + (size_t)(m0 + l) * D + 2 * half; // K base per half
    const float* Bcol = W + (size_t)(2 * half) * D + (j0 + l);

    v8f c0 = {};
    v8f c1 = {};
    #pragma unroll 4
    for (int k = 0; k < D; k += 8) {
        // even K-quad -> chain 0
        {
            v2f a;
            a.x = Arow[k];
            a.y = Arow[k + 1];
            v2f bb;
            bb.x = Bcol[(size_t)k * D];
            bb.y = Bcol[(size_t)(k + 1) * D];
            c0 = __builtin_amdgcn_wmma_f32_16x16x4_f32(
                     false, a, false, bb, (short)0, c0, false, false);
        }
        // odd K-quad -> chain 1 (independent of chain 0)
        {
            v2f a;
            a.x = Arow[k + 4];
            a.y = Arow[k + 5];
            v2f bb;
            bb.x = Bcol[(size_t)(k + 4) * D];
            bb.y = Bcol[(size_t)(k + 5) * D];
            c1 = __builtin_amdgcn_wmma_f32_16x16x4_f32(
                     false, a, false, bb, (short)0, c1, false, false);
        }
    }

    const float bj = bias[j0 + l];
    float lsum = 0.0f, lsq = 0.0f;
    #pragma unroll
    for (int v = 0; v < 8; ++v) {
        float h = c0[v] + c1[v] + bj;
        h = (h >= 0.0f) ? h : a1 * h;          // PReLU(a1)
        h_out[(size_t)(m0 + v + 8 * half) * D + (j0 + l)] = h;
        lsum += h;
        lsq  += h * h;
    }

    // Per-column reduction: LDS first (2 contributors per column per block),
    // then one global atomic per column per block.
    atomicAdd(&s_sum[j0 + l], lsum);
    atomicAdd(&s_sq[j0 + l], lsq);
    __syncthreads();
    if (tid < D) {
        atomicAdd(&colsum[tid],  s_sum[tid]);
        atomicAdd(&colsumsq[tid], s_sq[tid]);
    }
}

// ---------------------------------------------------------------------------
// Kernel 3: fold BN stats + gamma/beta into per-column scale/shift.
// ---------------------------------------------------------------------------
__global__ void bn_finalize(const float* __restrict__ colsum,
                            const float* __restrict__ colsumsq,
                            const float* __restrict__ gamma,
                            const float* __restrict__ beta,
                            float* __restrict__ scale,
                            float* __restrict__ shift, int N)
{
    int j = threadIdx.x;  // 128 threads
    float invN = 1.0f / (float)N;
    float m   = colsum[j] * invN;
    float var = colsumsq[j] * invN - m * m;
    float inv = rsqrtf(var + 1e-5f);
    float sc  = gamma[j] * inv;
    scale[j] = sc;
    shift[j] = beta[j] - m * sc;
}

// ---------------------------------------------------------------------------
// Kernel 4: in-place BN apply + PReLU(a2) on d_out, float4 vectorized.
// ---------------------------------------------------------------------------
__global__ __launch_bounds__(256) void bn_prelu(
    float* __restrict__ h, const float* __restrict__ scale,
    const float* __restrict__ shift, const float* __restrict__ a2p,
    size_t nquads)
{
    size_t i = (size_t)blockIdx.x * blockDim.x + threadIdx.x;
    if (i >= nquads) return;
    const float a2 = a2p[0];
    int j = (int)((i * 4) & (D - 1));  // column base (D divisible by 4)

    float4 v = ((float4*)h)[i];
    float x;
    x = v.x * scale[j + 0] + shift[j + 0]; v.x = (x >= 0.0f) ? x : a2 * x;
    x = v.y * scale[j + 1] + shift[j + 1]; v.y = (x >= 0.0f) ? x : a2 * x;
    x = v.z * scale[j + 2] + shift[j + 2]; v.z = (x >= 0.0f) ? x : a2 * x;
    x = v.w * scale[j + 3] + shift[j + 3]; v.w = (x >= 0.0f) ? x : a2 * x;
    ((float4*)h)[i] = v;
}

// ---------------------------------------------------------------------------
// Launch
// ---------------------------------------------------------------------------
extern "C" void kernel_launch(void* const* d_in, const int* in_sizes, int n_in,
                              void* d_out, int out_size, void* d_ws, size_t ws_size,
                              hipStream_t stream) {
    const float* feat  = (const float*)d_in[0];
    const int*   src   = (const int*)d_in[1];
    const int*   dst   = (const int*)d_in[2];
    const float* ew    = (const float*)d_in[3];
    const float* Wm    = (const float*)d_in[4];
    const float* bias  = (const float*)d_in[5];
    const float* a1    = (const float*)d_in[6];
    const float* gamma = (const float*)d_in[7];
    const float* beta  = (const float*)d_in[8];
    const float* a2    = (const float*)d_in[9];

    const int N = in_sizes[0] / D;   // 100000
    const int E = in_sizes[1];       // 1600000
    float* out = (float*)d_out;      // doubles as the h buffer

    // Workspace layout: agg [N*D] | colsum[128] | colsumsq[128] | scale[128] | shift[128]
    float* agg      = (float*)d_ws;
    float* colsum   = agg + (size_t)N * D;
    float* colsumsq = colsum + D;
    float* scale    = colsumsq + D;
    float* shift    = scale + D;

    // 0) zero agg + colsum/colsumsq
    size_t nz = (size_t)N * D + 2 * D;
    zero_f32<<<2048, 256, 0, stream>>>(agg, nz);

    // 1) edge gather-scale-scatter (1 wave per edge)
    long long sthreads = (long long)E * 32;
    int sblocks = (int)((sthreads + 255) / 256);
    edge_scatter<<<sblocks, 256, 0, stream>>>(feat, src, dst, ew, agg, E);

    // 2) WMMA GEMM + bias + PReLU(a1) + BN partial sums (N divisible by 16)
    gemm_wmma<<<N / 16, 256, 0, stream>>>(agg, Wm, bias, a1, out, colsum, colsumsq);

    // 3) BN stat finalize
    bn_finalize<<<1, D, 0, stream>>>(colsum, colsumsq, gamma, beta, scale, shift, N);

    // 4) BN apply + PReLU(a2), in place on d_out
    size_t nquads = (size_t)N * D / 4;
    int qblocks = (int)((nquads + 255) / 256);
    bn_prelu<<<qblocks, 256, 0, stream>>>(out, scale, shift, a2, nquads);
}